// HardNegLoss_15857019257550
// MI455X (gfx1250) — compile-verified
//
#include <hip/hip_runtime.h>
#include <stdint.h>

// HardNegLoss for MI455X (gfx1250, wave32).
// Design: one workgroup per row. Stage pred-row + target-row (20 KB each) into
// LDS once via the Tensor Data Mover (async DMA, TENSORcnt), then:
//   pass 1: pos count + positive BCE   (softplus(x) - x over target==1)
//   pass 2: 4x 256-bin radix-select of the k-th largest negative logit
//   pass 3: sum softplus(x) over selected negatives (+ stable tie handling)
// Global traffic = one read of each input (~328 MB) -> ~14 us floor at 23.3 TB/s.
// Softmax is provably unnecessary: ranking negatives by prob == ranking by logit.

#define N_ROWS 8192
#define C_COLS 5000
#define RATIO  3
#define BLOCK  256
#define TIE_CAP 32

typedef uint32_t v4u __attribute__((ext_vector_type(4)));
typedef int32_t  v8i __attribute__((ext_vector_type(8)));
typedef int32_t  v4i __attribute__((ext_vector_type(4)));

__device__ __forceinline__ unsigned fkey(float x) {
    // order-preserving float -> uint map (ascending)
    unsigned u = __float_as_uint(x);
    return (u & 0x80000000u) ? ~u : (u | 0x80000000u);
}

__device__ __forceinline__ float softplusf(float x) {
    return fmaxf(x, 0.0f) + log1pf(__expf(-fabsf(x)));
}

__device__ __forceinline__ float block_reduce_sum(float v, float* s_red) {
    const int tid = threadIdx.x;
    s_red[tid] = v;
    __syncthreads();
    #pragma unroll
    for (int off = BLOCK / 2; off > 0; off >>= 1) {
        if (tid < off) s_red[tid] += s_red[tid + off];
        __syncthreads();
    }
    float r = s_red[0];
    __syncthreads();
    return r;
}

// Issue a TDM DMA of `nelem` f32 from global to LDS. 1-row 2D descriptor.
// amdgpu-toolchain (clang-23) 6-arg builtin:
//   (uint32x4 g0, int32x8 g1, int32x4 g2, int32x4 g3, int32x8 g4, i32 cpol)
__device__ __forceinline__ void tdm_load_row_f32(const float* gsrc,
                                                 uint32_t lds_byte_addr,
                                                 int nelem) {
    uint64_t ga = (uint64_t)(uintptr_t)gsrc;
    v4u g0;
    g0.x = 1u;                                    // count=1 valid, user mode
    g0.y = lds_byte_addr;                         // D#.lds_addr
    g0.z = (uint32_t)ga;                          // global_addr[31:0]
    g0.w = (uint32_t)((ga >> 32) & 0x01FFFFFFull) // global_addr[56:32]
         | (2u << 30);                            // type=2 ("image")
    v8i g1;
    g1[0] = 0x00020000;                                   // data_size=2 (4B), no flags
    g1[1] = (int)(((uint32_t)nelem & 0xFFFFu) << 16);     // tensor_dim0[15:0]
    g1[2] = (int)((1u << 16) |                            // tensor_dim1 = 1
                  (((uint32_t)nelem >> 16) & 0xFFFFu));   // tensor_dim0[31:16]
    g1[3] = (int)(((uint32_t)nelem & 0xFFFFu) << 16);     // tile_dim0 = nelem
    g1[4] = 1;                                            // tile_dim1 = 1, tile_dim2 = 0
    g1[5] = nelem;                                        // tensor_dim0_stride[31:0]
    g1[6] = 0;                                            // stride hi / dim1_stride lo
    g1[7] = 0;
    v4i z4 = {0, 0, 0, 0};
    v8i z8 = {0, 0, 0, 0, 0, 0, 0, 0};
    __builtin_amdgcn_tensor_load_to_lds(g0, g1, z4, z4, z8, 0);
}

__global__ __launch_bounds__(BLOCK)
void hardneg_rows(const float* __restrict__ pred,
                  const float* __restrict__ tgt,
                  float* __restrict__ acc /* acc[0]=loss sum, acc[1]=pos sum */) {
    __shared__ __align__(16) float s_pred[C_COLS];
    __shared__ __align__(16) float s_tgt[C_COLS];
    __shared__ float    s_red[BLOCK];
    __shared__ unsigned s_hist[256];
    __shared__ unsigned s_prefix;
    __shared__ int      s_krem;
    __shared__ int      s_tcnt;
    __shared__ int      s_tidx[TIE_CAP];

    const int tid = threadIdx.x;
    const int row = blockIdx.x;
    const float* prow = pred + (size_t)row * C_COLS;
    const float* trow = tgt  + (size_t)row * C_COLS;

    // ---- Stage both rows into LDS via TDM (wave 0 issues, all waves consume) ----
    if (tid < 32) {
        tdm_load_row_f32(prow, (uint32_t)(uintptr_t)&s_pred[0], C_COLS);
        tdm_load_row_f32(trow, (uint32_t)(uintptr_t)&s_tgt[0],  C_COLS);
        __builtin_amdgcn_s_wait_tensorcnt(0);
    }
    __syncthreads();

    // ---- Pass 1: positives count + positive-class BCE ----
    float posv = 0.0f, bcep = 0.0f;
    for (int i = tid; i < C_COLS; i += BLOCK) {
        float t = s_tgt[i];
        if (t > 0.5f) {
            float x = s_pred[i];
            posv += 1.0f;
            bcep += softplusf(x) - x;   // softplus(x) - 1*x
        }
    }
    const float pos     = block_reduce_sum(posv, s_red);
    const float bce_pos = block_reduce_sum(bcep, s_red);
    const int   posi    = (int)(pos + 0.5f);
    const int   k       = min(RATIO * posi, C_COLS - posi);

    float row_loss_extra = 0.0f;   // only meaningful in tid 0

    if (k > 0) {
        // ---- Pass 2: radix-select k-th largest key among negatives ----
        if (tid == 0) { s_prefix = 0u; s_krem = k; }
        __syncthreads();
        for (int p = 0; p < 4; ++p) {
            const int shift = 24 - 8 * p;
            s_hist[tid] = 0u;
            __syncthreads();
            const unsigned pref = s_prefix;
            for (int i = tid; i < C_COLS; i += BLOCK) {
                if (s_tgt[i] <= 0.5f) {
                    unsigned key = fkey(s_pred[i]);
                    bool cand = (p == 0) || ((key >> (shift + 8)) == pref);
                    if (cand) atomicAdd(&s_hist[(key >> shift) & 0xFFu], 1u);
                }
            }
            __syncthreads();
            if (tid == 0) {
                const unsigned krem = (unsigned)s_krem;
                unsigned cum = 0;
                int b = 255;
                for (; b > 0; --b) {
                    unsigned h = s_hist[b];
                    if (cum + h >= krem) break;
                    cum += h;
                }
                s_prefix = (pref << 8) | (unsigned)b;
                s_krem   = (int)(krem - cum);
            }
            __syncthreads();
        }
        const unsigned T  = s_prefix;   // key of the k-th largest negative
        const int needed  = s_krem;     // how many key==T entries to take

        // ---- Pass 3: sum softplus over selected negatives ----
        float acc_gt = 0.0f;
        if (tid == 0) s_tcnt = 0;
        __syncthreads();
        for (int i = tid; i < C_COLS; i += BLOCK) {
            if (s_tgt[i] <= 0.5f) {
                unsigned key = fkey(s_pred[i]);
                if (key > T) {
                    acc_gt += softplusf(s_pred[i]);
                } else if (key == T) {
                    int slot = atomicAdd(&s_tcnt, 1);
                    if (slot < TIE_CAP) s_tidx[slot] = i;
                }
            }
        }
        const float sum_gt = block_reduce_sum(acc_gt, s_red);
        if (tid == 0) {
            // stable tie-break: lowest original index first (matches jnp stable argsort)
            int cnt = min(s_tcnt, TIE_CAP);
            for (int a = 1; a < cnt; ++a) {
                int v = s_tidx[a]; int b2 = a - 1;
                while (b2 >= 0 && s_tidx[b2] > v) { s_tidx[b2 + 1] = s_tidx[b2]; --b2; }
                s_tidx[b2 + 1] = v;
            }
            int take = min(needed, cnt);
            float ts = 0.0f;
            for (int j = 0; j < take; ++j) ts += softplusf(s_pred[s_tidx[j]]);
            row_loss_extra = sum_gt + ts;
        }
    }

    if (tid == 0) {
        atomicAdd(&acc[0], bce_pos + row_loss_extra);
        atomicAdd(&acc[1], pos);
    }
}

__global__ void hardneg_init(float* acc) {
    if (threadIdx.x < 2) acc[threadIdx.x] = 0.0f;
}

__global__ void hardneg_finalize(const float* __restrict__ acc, float* __restrict__ out) {
    if (threadIdx.x == 0) out[0] = acc[0] / acc[1];
}

extern "C" void kernel_launch(void* const* d_in, const int* in_sizes, int n_in,
                              void* d_out, int out_size, void* d_ws, size_t ws_size,
                              hipStream_t stream) {
    const float* pred = (const float*)d_in[0];
    const float* tgt  = (const float*)d_in[1];
    float* acc = (float*)d_ws;     // acc[0] = loss sum, acc[1] = positives sum
    float* out = (float*)d_out;

    hardneg_init<<<1, 32, 0, stream>>>(acc);
    hardneg_rows<<<N_ROWS, BLOCK, 0, stream>>>(pred, tgt, acc);
    hardneg_finalize<<<1, 32, 0, stream>>>(acc, out);
}